// RNNCell_75685913690463
// MI455X (gfx1250) — compile-verified
//
#include <hip/hip_runtime.h>

typedef __attribute__((ext_vector_type(2))) float v2f;
typedef __attribute__((ext_vector_type(4))) float v4f;
typedef __attribute__((ext_vector_type(8))) float v8f;

#define Hn 512
#define In 512

// ---------------------------------------------------------------------------
// Kernel 1: a = [W_ih | W_hh] @ [x; h] + b via V_WMMA_F32_16X16X4_F32.
// One block of 1024 threads = 32 waves; wave w computes rows [16w, 16w+16).
// A tile (16x4 f32, 2 VGPRs/lane): lane r (0-15) holds A[r][k0],A[r][k0+1]
// with k0 = 2*(lane>>4) relative K. B (4x16): vector chunk in column N=0.
// D column 0 lands in lane 0 (rows 0..7, acc[0..7]) and lane 16 (rows 8..15).
// ---------------------------------------------------------------------------
__global__ __launch_bounds__(1024) void rnn_matvec_wmma(
    const float* __restrict__ x, const float* __restrict__ h,
    const float* __restrict__ Wih, const float* __restrict__ Whh,
    const float* __restrict__ bias,
    float* __restrict__ out_new0, float* __restrict__ out_new1,
    float* __restrict__ d_vec)
{
    const int wave = threadIdx.x >> 5;   // 0..31  -> row tile
    const int lane = threadIdx.x & 31;
    const int r    = lane & 15;
    const int half = lane >> 4;          // 0 or 1
    const int rowBase = wave * 16;
    const int row     = rowBase + r;

    v8f acc = {};

    // K in [0, 512): W_ih and x
    for (int k = 0; k < In; k += 4) {
        const int k0 = k + 2 * half;
        v2f A = { Wih[row * In + k0], Wih[row * In + k0 + 1] };
        v2f B = { 0.0f, 0.0f };
        if (r == 0) { B.x = x[k0]; B.y = x[k0 + 1]; }
        acc = __builtin_amdgcn_wmma_f32_16x16x4_f32(
            false, A, false, B, (short)0, acc, false, false);
    }
    // K in [512, 1024): W_hh and h
    for (int k = 0; k < Hn; k += 4) {
        const int k0 = k + 2 * half;
        v2f A = { Whh[row * Hn + k0], Whh[row * Hn + k0 + 1] };
        v2f B = { 0.0f, 0.0f };
        if (r == 0) { B.x = h[k0]; B.y = h[k0 + 1]; }
        acc = __builtin_amdgcn_wmma_f32_16x16x4_f32(
            false, A, false, B, (short)0, acc, false, false);
    }

    // Column N=0 of D: lane 0 -> rows rowBase+0..7, lane 16 -> rowBase+8..15
    if (r == 0) {
        #pragma unroll
        for (int m = 0; m < 8; ++m) {
            const int orow = rowBase + half * 8 + m;
            const float a  = acc[m] + bias[orow];
            const float t  = tanhf(a);
            const float nw = (t > 0.0f) ? 1.0f : 0.0f;  // heaviside(t, t)
            const float dd = 1.0f - t * t;
            out_new0[orow] = nw;
            out_new1[orow] = nw;
            d_vec[orow]    = dd;
        }
    }
}

// ---------------------------------------------------------------------------
// Kernel 2: stream 512 MB of zeros into bar_M with non-temporal b128 stores.
// Output (538 MB) >> L2 (192 MB) and is never re-read on device -> NT.
// ---------------------------------------------------------------------------
__global__ __launch_bounds__(256) void barM_zero(v4f* __restrict__ p, size_t n4)
{
    size_t i      = (size_t)blockIdx.x * blockDim.x + threadIdx.x;
    const size_t stride = (size_t)gridDim.x * blockDim.x;
    const v4f z = {};
    for (; i < n4; i += stride)
        __builtin_nontemporal_store(z, p + i);
}

// ---------------------------------------------------------------------------
// Kernel 3: overwrite the block-diagonal rows of bar_M with d_i * h_k and
// produce J = d[:,None] * W_hh. 262144 lanes; ~3 MB of traffic.
// ---------------------------------------------------------------------------
__global__ __launch_bounds__(256) void barM_diag_J(
    const float* __restrict__ d_vec, const float* __restrict__ h,
    const float* __restrict__ Whh,
    float* __restrict__ barM, float* __restrict__ J)
{
    const int idx = blockIdx.x * blockDim.x + threadIdx.x;   // 0 .. 512*512-1
    const int i = idx >> 9;      // row of J / diagonal index
    const int k = idx & 511;     // column
    const float di = d_vec[i];
    // bar_M[i, i, k] = d_i * h_k   (flat offset i*512*512 + i*512 + k)
    barM[(size_t)i * (size_t)(Hn * Hn + Hn) + k] = di * h[k];
    J[idx] = di * Whh[idx];
}

extern "C" void kernel_launch(void* const* d_in, const int* in_sizes, int n_in,
                              void* d_out, int out_size, void* d_ws, size_t ws_size,
                              hipStream_t stream) {
    const float* x    = (const float*)d_in[0];  // input   (512)
    const float* h    = (const float*)d_in[1];  // hidden  (512)
    const float* Wih  = (const float*)d_in[2];  // (512,512)
    const float* Whh  = (const float*)d_in[3];  // (512,512)
    const float* bias = (const float*)d_in[4];  // (512)

    float* out      = (float*)d_out;
    float* out_new0 = out;                            // new (512)
    float* out_new1 = out + Hn;                       // new again (512)
    float* barM     = out + 2 * Hn;                   // (512,512,512)
    float* J        = barM + (size_t)Hn * Hn * Hn;    // (512,512)

    float* d_vec = (float*)d_ws;                      // 512 floats scratch

    // 1) tiny WMMA matvec -> new, d
    rnn_matvec_wmma<<<1, 1024, 0, stream>>>(x, h, Wih, Whh, bias,
                                            out_new0, out_new1, d_vec);

    // 2) 512 MB NT zero-fill of bar_M (the roofline-dominant step)
    const size_t n4 = ((size_t)Hn * Hn * Hn) / 4;     // 33,554,432 float4s
    barM_zero<<<4096, 256, 0, stream>>>((v4f*)barM, n4);

    // 3) diagonal of bar_M + J
    barM_diag_J<<<(Hn * Hn) / 256, 256, 0, stream>>>(d_vec, h, Whh, barM, J);
}